// Graph_SMOTE_37958920962738
// MI455X (gfx1250) — compile-verified
//
#include <hip/hip_runtime.h>

// GraphSAGE 2-layer inference for MI455X (gfx1250, wave32, WMMA).
// Aggregation is bandwidth-bound (~3.3 GB @ 23.3 TB/s); GEMMs use
// V_WMMA_F32_16X16X4_F32 to keep full f32 precision (compute is not the
// bottleneck, so the f32 matrix path is the right choice over bf16).

typedef float v2f __attribute__((ext_vector_type(2)));
typedef float v8f __attribute__((ext_vector_type(8)));

#define DF 128          // D_IN == D_HID == 128
#define DOUT 16
#define LDS_PITCH 130   // pad 128 -> 130 (stride of 2 banks across lanes)

// ---------------------------------------------------------------- utilities
__global__ void zero_f32(float* __restrict__ p, long n) {
  long i = (long)blockIdx.x * blockDim.x + threadIdx.x;
  long stride = (long)gridDim.x * blockDim.x;
  for (; i < n; i += stride) p[i] = 0.0f;
}

// degree count (float so the divide path is branch-free later)
__global__ void degree_kernel(const int* __restrict__ dst, float* __restrict__ cnt, int E) {
  int e = blockIdx.x * blockDim.x + threadIdx.x;
  if (e < E) atomicAdd(&cnt[dst[e]], 1.0f);
}

// one wave (32 lanes) per edge; each lane moves 4 contiguous floats.
// float4 gather (coalesced per-edge) + 4 global f32 atomics scatter.
__global__ void scatter_kernel(const float* __restrict__ x,
                               const int* __restrict__ src,
                               const int* __restrict__ dst,
                               float* __restrict__ sum, int E) {
  long t = (long)blockIdx.x * blockDim.x + threadIdx.x;
  int e = (int)(t >> 5);
  int lane = (int)(t & 31);
  if (e >= E) return;
  int s = src[e], d = dst[e];
  const float4* xin = (const float4*)(x + (long)s * DF);
  float4 v = xin[lane];
  float* o = sum + (long)d * DF + lane * 4;
  atomicAdd(o + 0, v.x);
  atomicAdd(o + 1, v.y);
  atomicAdd(o + 2, v.z);
  atomicAdd(o + 3, v.w);
}

// ------------------------------------------------ fused SAGE layer GEMM
// out[node] = relu?( agg[node] @ Wl^T + x[node] @ Wr^T + b )
// Block = 256 threads = 8 waves; block owns 16 nodes x 128 out-features.
// Wave w computes the 16x16 tile at out-cols [16w, 16w+16), K=128 via
// 32 k-steps x 2 chained V_WMMA_F32_16X16X4_F32 (one per weight matrix).
__global__ void __launch_bounds__(256)
sage_gemm(const float* __restrict__ sum, const float* __restrict__ cnt,
          const float* __restrict__ xin,
          const float* __restrict__ Wl, const float* __restrict__ bl,
          const float* __restrict__ Wr,
          float* __restrict__ out, int N, int relu) {
  __shared__ float lds_a[16 * LDS_PITCH];  // mean-aggregated neighbor feats
  __shared__ float lds_x[16 * LDS_PITCH];  // self feats
  const int base = blockIdx.x * 16;
  const int tid = threadIdx.x;

  // cooperative load of the 16x128 A-tiles (zero-pad past N)
  for (int i = tid; i < 16 * DF; i += 256) {
    int r = i >> 7;        // node row in tile
    int c = i & (DF - 1);  // feature col
    int node = base + r;
    float a = 0.0f, xv = 0.0f;
    if (node < N) {
      float inv = 1.0f / fmaxf(cnt[node], 1.0f);
      a  = sum[(long)node * DF + c] * inv;
      xv = xin[(long)node * DF + c];
    }
    lds_a[r * LDS_PITCH + c] = a;
    lds_x[r * LDS_PITCH + c] = xv;
  }
  __syncthreads();

  const int wave = tid >> 5;
  const int lane = tid & 31;
  const int hv = lane >> 4;   // 0: holds K=k0..k0+1   1: holds K=k0+2..k0+3
  const int lm = lane & 15;   // M (for A) / N (for B) within the tile
  const int n = wave * 16 + lm;  // global output feature

  const float* wl_row = Wl + (long)n * DF;  // B[k][n] = W[n][k]
  const float* wr_row = Wr + (long)n * DF;

  v8f c = {};
  for (int k0 = 0; k0 < DF; k0 += 4) {
    int kk = k0 + 2 * hv;
    v2f aa = *(const v2f*)&lds_a[lm * LDS_PITCH + kk];
    v2f ax = *(const v2f*)&lds_x[lm * LDS_PITCH + kk];
    v2f bl2 = *(const v2f*)&wl_row[kk];
    v2f br2 = *(const v2f*)&wr_row[kk];
    c = __builtin_amdgcn_wmma_f32_16x16x4_f32(false, aa, false, bl2,
                                              (short)0, c, false, false);
    c = __builtin_amdgcn_wmma_f32_16x16x4_f32(false, ax, false, br2,
                                              (short)0, c, false, false);
  }

  // C/D layout: VGPR v -> M = v + 8*hv, N = lm
  float bias = bl[n];
  for (int v = 0; v < 8; v++) {
    int node = base + v + 8 * hv;
    if (node < N) {
      float val = c[v] + bias;
      if (relu) val = fmaxf(val, 0.0f);
      out[(long)node * DF + n] = val;
    }
  }
}

// ------------------------------------------------ classifier: z @ Wc^T + bc
// One wave per 16-node tile (D_OUT = 16 -> a single 16x16 WMMA tile, K=128).
__global__ void __launch_bounds__(256)
classifier_gemm(const float* __restrict__ z, const float* __restrict__ Wc,
                const float* __restrict__ bc, float* __restrict__ out, int N) {
  const int tid = threadIdx.x;
  const int wave = tid >> 5, lane = tid & 31;
  const int hv = lane >> 4, lm = lane & 15;
  const int base = blockIdx.x * 128 + wave * 16;

  int nodeA = base + lm;
  int nodeAc = nodeA < N ? nodeA : (N - 1);  // clamp: keep EXEC all-ones
  const float* zrow = z + (long)nodeAc * DF;
  const float* wrow = Wc + (long)lm * DF;    // B[k][n] = Wc[n][k], n = lm

  v8f c = {};
  for (int k0 = 0; k0 < DF; k0 += 4) {
    int kk = k0 + 2 * hv;
    v2f aa = *(const v2f*)&zrow[kk];
    v2f bb = *(const v2f*)&wrow[kk];
    c = __builtin_amdgcn_wmma_f32_16x16x4_f32(false, aa, false, bb,
                                              (short)0, c, false, false);
  }

  float bias = bc[lm];
  for (int v = 0; v < 8; v++) {
    int node = base + v + 8 * hv;
    if (node < N) out[(long)node * DOUT + lm] = c[v] + bias;
  }
}

// ---------------------------------------------------------------- launcher
extern "C" void kernel_launch(void* const* d_in, const int* in_sizes, int n_in,
                              void* d_out, int out_size, void* d_ws, size_t ws_size,
                              hipStream_t stream) {
  const float* x   = (const float*)d_in[0];
  const int*   ei  = (const int*)  d_in[1];
  const float* W1l = (const float*)d_in[2];
  const float* b1l = (const float*)d_in[3];
  const float* W1r = (const float*)d_in[4];
  const float* W2l = (const float*)d_in[5];
  const float* b2l = (const float*)d_in[6];
  const float* W2r = (const float*)d_in[7];
  const float* Wc  = (const float*)d_in[8];
  const float* bc  = (const float*)d_in[9];
  float* out = (float*)d_out;

  const int N = in_sizes[0] / DF;
  const int E = in_sizes[1] / 2;
  const int* src = ei;       // edge_index[0]
  const int* dst = ei + E;   // edge_index[1]

  // workspace layout (floats): cnt[N] | sum[N*128] | h[N*128] | z[N*128]
  float* ws   = (float*)d_ws;
  float* cnt  = ws;
  float* sumb = ws + N;
  float* h    = sumb + (long)N * DF;
  float* zbuf = h    + (long)N * DF;

  const int sblocks = (int)(((long)E * 32 + 255) / 256);
  const int gblocks = (N + 15) / 16;

  // layer 1
  zero_f32<<<2048, 256, 0, stream>>>(cnt, (long)N * DF + N);  // cnt+sum contiguous
  degree_kernel<<<(E + 255) / 256, 256, 0, stream>>>(dst, cnt, E);
  scatter_kernel<<<sblocks, 256, 0, stream>>>(x, src, dst, sumb, E);
  sage_gemm<<<gblocks, 256, 0, stream>>>(sumb, cnt, x, W1l, b1l, W1r, h, N, 1);

  // layer 2 (dropout is identity at inference)
  zero_f32<<<2048, 256, 0, stream>>>(sumb, (long)N * DF);
  scatter_kernel<<<sblocks, 256, 0, stream>>>(h, src, dst, sumb, E);
  sage_gemm<<<gblocks, 256, 0, stream>>>(sumb, cnt, h, W2l, b2l, W2r, zbuf, N, 0);

  // classifier
  classifier_gemm<<<(N + 127) / 128, 256, 0, stream>>>(zbuf, Wc, bc, out, N);
}